// BikeSafetyGNN_5042291606016
// MI455X (gfx1250) — compile-verified
//
#include <hip/hip_runtime.h>

typedef __attribute__((ext_vector_type(2))) float v2f;
typedef __attribute__((ext_vector_type(8))) float v8f;

// ---------------- utility kernels ----------------

__global__ void zero_f32(float* __restrict__ p, long n) {
  long i = (long)blockIdx.x * blockDim.x + threadIdx.x;
  long stride = (long)gridDim.x * blockDim.x;
  for (; i < n; i += stride) p[i] = 0.0f;
}

__global__ void count_edges(const long long* __restrict__ dst, long E, float* __restrict__ cnt) {
  long e = (long)blockIdx.x * blockDim.x + threadIdx.x;
  if (e >= E) return;
  unsafeAtomicAdd(&cnt[dst[e]], 1.0f);   // global_atomic_add_f32
}

__global__ void finalize_inv(float* __restrict__ c, long n) {
  long i = (long)blockIdx.x * blockDim.x + threadIdx.x;
  if (i < n) c[i] = 1.0f / fmaxf(c[i], 1.0f);
}

// ---------------- edge aggregation: s[dst] += h[src] ----------------
// One thread per (edge, 4-float chunk). For DIN=128 a whole wave shares one
// edge (wave-uniform index load), chunks are consecutive -> coalesced float4
// gathers; scatter uses hardware fp32 atomics (L2-resident tables).

template<int DIN>
__global__ void aggregate(const float* __restrict__ h,
                          const long long* __restrict__ src,
                          const long long* __restrict__ dst,
                          long E, float* __restrict__ s) {
  constexpr int CH = DIN / 4;
  long t = (long)blockIdx.x * blockDim.x + threadIdx.x;
  if (t >= E * (long)CH) return;
  long e = t / CH;
  int  c = (int)(t - e * CH) * 4;
  long sn = (long)src[e];
  long dn = (long)dst[e];
  const float4 v = *(const float4*)(h + sn * DIN + c);
  float* p = s + dn * DIN + c;
  unsafeAtomicAdd(p + 0, v.x);
  unsafeAtomicAdd(p + 1, v.y);
  unsafeAtomicAdd(p + 2, v.z);
  unsafeAtomicAdd(p + 3, v.w);
}

// ---------------- fused SAGE layer GEMM ----------------
// out = relu( (s*invc) @ Wl^T + bl + hin @ Wr^T )  as one WMMA chain over
// K = 2*DIN using V_WMMA_F32_16X16X4_F32. One block = 16 nodes, one wave per
// 16 output columns.

template<int DIN, int DOUT>
__global__ void sage_gemm(const float* __restrict__ s, const float* __restrict__ invc,
                          const float* __restrict__ hin,
                          const float* __restrict__ Wl, const float* __restrict__ bl,
                          const float* __restrict__ Wr,
                          float* __restrict__ out, long N) {
  constexpr int LSTR = DIN + 4;                 // bank-conflict-free LDS stride
  __shared__ float Am[16 * LSTR];               // mean rows
  __shared__ float Ah[16 * LSTR];               // raw feature rows
  const long base = (long)blockIdx.x * 16;
  const int tid = threadIdx.x;
  const int nth = blockDim.x;

  // cooperative staging: 16 rows x DIN floats, mean scaled by invc
  for (int i = tid; i < 16 * (DIN / 4); i += nth) {
    int row = i / (DIN / 4);
    int c4  = (i % (DIN / 4)) * 4;
    long node = base + row; if (node >= N) node = N - 1;
    float inv = invc[node];
    float4 sv = *(const float4*)(s   + node * DIN + c4);
    float4 hv = *(const float4*)(hin + node * DIN + c4);
    float* am = &Am[row * LSTR + c4];
    am[0] = sv.x * inv; am[1] = sv.y * inv; am[2] = sv.z * inv; am[3] = sv.w * inv;
    *(float4*)&Ah[row * LSTR + c4] = hv;
  }
  __syncthreads();

  const int lane = tid & 31;
  const int wave = tid >> 5;
  const int n0   = wave * 16;              // this wave's output-column tile
  const int m    = lane & 15;
  const int kh   = (lane >> 4) << 1;       // 0 for lanes 0-15, 2 for 16-31

  const float* wl = Wl + (long)(n0 + m) * DIN;  // B[k][n] = W[n][k]
  const float* wr = Wr + (long)(n0 + m) * DIN;
  const float* am = &Am[m * LSTR];
  const float* ah = &Ah[m * LSTR];

  v8f acc = {0.f, 0.f, 0.f, 0.f, 0.f, 0.f, 0.f, 0.f};

#pragma unroll
  for (int k = 0; k < DIN; k += 4) {       // mean @ Wl^T
    v2f a, b;
    a.x = am[k + kh];  a.y = am[k + kh + 1];
    b.x = wl[k + kh];  b.y = wl[k + kh + 1];
    acc = __builtin_amdgcn_wmma_f32_16x16x4_f32(false, a, false, b,
                                                (short)0, acc, false, false);
  }
#pragma unroll
  for (int k = 0; k < DIN; k += 4) {       // + hin @ Wr^T (same accumulator)
    v2f a, b;
    a.x = ah[k + kh];  a.y = ah[k + kh + 1];
    b.x = wr[k + kh];  b.y = wr[k + kh + 1];
    acc = __builtin_amdgcn_wmma_f32_16x16x4_f32(false, a, false, b,
                                                (short)0, acc, false, false);
  }

  // epilogue: bias + relu + store. C/D layout: VGPR r -> M=r (lanes 0-15),
  // M=r+8 (lanes 16-31); N = n0 + (lane&15).
  const float bias = bl[n0 + m];
  const int rb = (lane >> 4) * 8;
#pragma unroll
  for (int r = 0; r < 8; ++r) {
    long node = base + rb + r;
    if (node < N) {
      float v = acc[r] + bias;
      out[node * DOUT + n0 + m] = fmaxf(v, 0.0f);
    }
  }
}

// ---------------- output heads ----------------

__global__ void heads(const float* __restrict__ h3,
                      const float* __restrict__ Wreg, const float* __restrict__ breg,
                      const float* __restrict__ Wcls, const float* __restrict__ bcls,
                      float* __restrict__ out, long N) {
  long i = (long)blockIdx.x * blockDim.x + threadIdx.x;
  if (i >= N) return;
  const float* h = h3 + i * 16;
  float r = 0.f, c = 0.f;
#pragma unroll
  for (int j = 0; j < 16; ++j) {
    float v = h[j];
    r = fmaf(v, Wreg[j], r);
    c = fmaf(v, Wcls[j], c);
  }
  out[i]     = r + breg[0];
  out[N + i] = c + bcls[0];
}

// ---------------- launcher ----------------

extern "C" void kernel_launch(void* const* d_in, const int* in_sizes, int n_in,
                              void* d_out, int out_size, void* d_ws, size_t ws_size,
                              hipStream_t stream) {
  const float*     x    = (const float*)d_in[0];
  const long long* ei   = (const long long*)d_in[1];
  const float* W1l = (const float*)d_in[2];
  const float* b1l = (const float*)d_in[3];
  const float* W1r = (const float*)d_in[4];
  const float* W2l = (const float*)d_in[5];
  const float* b2l = (const float*)d_in[6];
  const float* W2r = (const float*)d_in[7];
  const float* W3l = (const float*)d_in[8];
  const float* b3l = (const float*)d_in[9];
  const float* W3r = (const float*)d_in[10];
  const float* Wreg = (const float*)d_in[11];
  const float* breg = (const float*)d_in[12];
  const float* Wcls = (const float*)d_in[13];
  const float* bcls = (const float*)d_in[14];

  const long N = (long)in_sizes[0] / 128;
  const long E = (long)in_sizes[1] / 2;
  const long long* srcp = ei;          // edge_index[0]
  const long long* dstp = ei + E;      // edge_index[1]

  // workspace layout (floats): invc[N] | A[N*128] | B[N*64]
  float* ws   = (float*)d_ws;
  float* invc = ws;
  float* A    = ws + N;
  float* B    = A + N * 128;
  float* s1 = A;             float* h1 = B;
  float* s2 = A;             float* h2 = A + N * 64;   // upper half of A
  float* s3 = B;             float* h3 = B + N * 32;   // h1 dead by then
  float* out = (float*)d_out;

  auto nb = [](long n, int t) { return (unsigned)((n + t - 1) / t); };
  const unsigned gtiles = nb(N, 16);

  // degree -> 1/max(deg,1)
  zero_f32<<<nb(N, 256), 256, 0, stream>>>(invc, N);
  count_edges<<<nb(E, 256), 256, 0, stream>>>(dstp, E, invc);
  finalize_inv<<<nb(N, 256), 256, 0, stream>>>(invc, N);

  // layer 1: 128 -> 64
  zero_f32<<<nb(N * 128, 256), 256, 0, stream>>>(s1, N * 128);
  aggregate<128><<<nb(E * 32, 256), 256, 0, stream>>>(x, srcp, dstp, E, s1);
  sage_gemm<128, 64><<<gtiles, 128, 0, stream>>>(s1, invc, x, W1l, b1l, W1r, h1, N);

  // layer 2: 64 -> 32
  zero_f32<<<nb(N * 64, 256), 256, 0, stream>>>(s2, N * 64);
  aggregate<64><<<nb(E * 16, 256), 256, 0, stream>>>(h1, srcp, dstp, E, s2);
  sage_gemm<64, 32><<<gtiles, 64, 0, stream>>>(s2, invc, h1, W2l, b2l, W2r, h2, N);

  // layer 3: 32 -> 16
  zero_f32<<<nb(N * 32, 256), 256, 0, stream>>>(s3, N * 32);
  aggregate<32><<<nb(E * 8, 256), 256, 0, stream>>>(h2, srcp, dstp, E, s3);
  sage_gemm<32, 16><<<gtiles, 32, 0, stream>>>(s3, invc, h2, W3l, b3l, W3r, h3, N);

  // heads -> (reg || cls)
  heads<<<nb(N, 256), 256, 0, stream>>>(h3, Wreg, breg, Wcls, bcls, out, N);
}